// LocalCorrWithGeom_88356067213328
// MI455X (gfx1250) — compile-verified
//
#include <hip/hip_runtime.h>
#include <cstdint>
#include <cstddef>

typedef __attribute__((ext_vector_type(2))) float v2f;
typedef __attribute__((ext_vector_type(8))) float v8f;

#define RWIN 3
#define DWIN 49
#define CIN  256
#define CMID 128
#define HDIM 112
#define WDIM 112
#define WT   (WDIM / 16)          // 7 tiles of 16 pixels per row
#define NPIX (2 * HDIM * WDIM)    // 25088 total pixels (B=2)

// ---------------------------------------------------------------------------
// Kernel 1: projection GEMM (Cmid x Cin) * (Cin x 16px) via V_WMMA_F32_16X16X4_F32
// followed by per-pixel L2 normalization. Output layout: [pix][CMID] (pixel-major).
// ---------------------------------------------------------------------------
__global__ __launch_bounds__(256)
void proj_norm_kernel(const float* __restrict__ phi,   // (B, CIN, H, W)
                      const float* __restrict__ Wm,    // (CMID, CIN) row-major
                      float* __restrict__ outbuf)      // (B*H*W, CMID)
{
    __shared__ float phi_s[CIN * 16];      // [c][px]   16 KB
    __shared__ float out_s[16 * CMID];     // [px][ch]   8 KB
    __shared__ float scale_s[16];

    const int t  = threadIdx.x;
    const int b  = blockIdx.y;
    const int hw = blockIdx.x;             // 0 .. H*WT-1
    const int h  = hw / WT;
    const int w0 = (hw % WT) * 16;

    // Stage the 256(ch) x 16(px) input tile into LDS, coalesced per channel row.
    {
        const float* pb = phi + (size_t)b * CIN * HDIM * WDIM + (size_t)h * WDIM + w0;
        const int px = t & 15;
        for (int c = t >> 4; c < CIN; c += 16)
            phi_s[c * 16 + px] = pb[(size_t)c * HDIM * WDIM + px];
    }
    __syncthreads();

    const int lane = t & 31;
    const int wv   = t >> 5;        // wave id 0..7 -> M-tile
    const int m0   = wv * 16;
    const int l    = lane & 15;
    const int hi   = lane >> 4;     // lane-half selects K-pair (ISA 16x4 f32 layout)

    v8f acc = {0.f, 0.f, 0.f, 0.f, 0.f, 0.f, 0.f, 0.f};
    const float* wbase = Wm + (size_t)(m0 + l) * CIN + 2 * hi;

    for (int kk = 0; kk < CIN; kk += 4) {
        // A fragment: row (m0+l), K = kk + 2*hi + {0,1}   (8B-aligned pair)
        v2f a = *(const v2f*)(wbase + kk);
        // B fragment: K = kk + 2*hi + {0,1}, N = l  (conflict-free [c][px] LDS)
        v2f bb;
        bb.x = phi_s[(kk + 2 * hi + 0) * 16 + l];
        bb.y = phi_s[(kk + 2 * hi + 1) * 16 + l];
        acc = __builtin_amdgcn_wmma_f32_16x16x4_f32(
            /*neg_a=*/false, a, /*neg_b=*/false, bb,
            /*c_mod=*/(short)0, acc, /*reuse_a=*/false, /*reuse_b=*/false);
    }

    // C/D layout: VGPR j -> M = m0 + j + 8*hi, N(pixel) = l
#pragma unroll
    for (int j = 0; j < 8; ++j)
        out_s[l * CMID + (m0 + j + 8 * hi)] = acc[j];
    __syncthreads();

    // Per-pixel L2 norm (16 pixels, one thread each)
    if (t < 16) {
        float s = 0.f;
        for (int c = 0; c < CMID; ++c) { float v = out_s[t * CMID + c]; s += v * v; }
        scale_s[t] = 1.0f / fmaxf(sqrtf(s), 1e-12f);
    }
    __syncthreads();

    // Scaled write-out, pixel-major: each thread writes 8 consecutive channels.
    {
        const int px = t >> 4;
        const int c0 = (t & 15) * 8;
        const size_t pix = ((size_t)b * HDIM + h) * WDIM + w0 + px;
        const float sc = scale_s[px];
        float* dst = outbuf + pix * CMID + c0;
#pragma unroll
        for (int j = 0; j < 8; ++j)
            dst[j] = out_s[px * CMID + c0 + j] * sc;
    }
}

// ---------------------------------------------------------------------------
// Kernel 2: fused correlation (49 x 128-dot), Wv matvec, geometry, softmax,
// expectation / confidence. 4 pixels per 256-thread block (64 threads/pixel).
// Zero-padding semantics match conv_general_dilated_patches exactly.
// ---------------------------------------------------------------------------
__global__ __launch_bounds__(256)
void corr_soft_kernel(const float* __restrict__ qbuf,  // (NPIX, CMID)
                      const float* __restrict__ kbuf,  // (NPIX, CMID)
                      const float* __restrict__ Pcur,  // (B, 3, H, W)
                      const float* __restrict__ Prnd,  // (B, 3, H, W)
                      const float* __restrict__ Wv,    // (49, 49)
                      const float* __restrict__ bv,    // (49)
                      const float* __restrict__ gamma_p,
                      float* __restrict__ out)         // [du | dv | conf], each NPIX
{
    __shared__ float q_s[4][CMID];
    __shared__ float corr_s[4][DWIN];
    __shared__ float geo_s[4][DWIN];
    __shared__ float lg_s[4][DWIN];
    __shared__ float mx_s[4];

    const int t   = threadIdx.x;
    const int i   = t >> 6;                 // local pixel 0..3
    const int r   = t & 63;                 // role within pixel
    const int pix = blockIdx.x * 4 + i;
    const int b   = pix / (HDIM * WDIM);
    const int rem = pix % (HDIM * WDIM);
    const int h   = rem / WDIM;
    const int w   = rem % WDIM;
    const float gamma = gamma_p[0];

    // Stage this pixel's q vector (512 B) via 32 float4 loads.
    if (r < 32) {
        const float4 v = *(const float4*)(qbuf + (size_t)pix * CMID + r * 4);
        q_s[i][r * 4 + 0] = v.x;
        q_s[i][r * 4 + 1] = v.y;
        q_s[i][r * 4 + 2] = v.z;
        q_s[i][r * 4 + 3] = v.w;
    }
    __syncthreads();

    // Phase B: lane d computes corr[d] and the geometric logit term.
    if (r < DWIN) {
        const int dy = r / 7 - RWIN;
        const int dx = r % 7 - RWIN;
        const int hn = h + dy, wn = w + dx;
        const bool inb = (hn >= 0) && (hn < HDIM) && (wn >= 0) && (wn < WDIM);

        float dot = 0.f, prx = 0.f, pry = 0.f, prz = 0.f;
        if (inb) {
            const size_t npix = ((size_t)b * HDIM + hn) * WDIM + wn;
            const float* kv = kbuf + npix * CMID;
            for (int c = 0; c < CMID; c += 4) {
                const float4 kc = *(const float4*)(kv + c);
                dot += q_s[i][c + 0] * kc.x + q_s[i][c + 1] * kc.y +
                       q_s[i][c + 2] * kc.z + q_s[i][c + 3] * kc.w;
            }
            const float* pr = Prnd + (size_t)b * 3 * HDIM * WDIM + (size_t)hn * WDIM + wn;
            prx = pr[0];
            pry = pr[HDIM * WDIM];
            prz = pr[2 * HDIM * WDIM];
        } // else: zero-padded patch -> dot = 0, Pr = 0

        const float* pc = Pcur + (size_t)b * 3 * HDIM * WDIM + (size_t)h * WDIM + w;
        const float dpx = pc[0] - prx;
        const float dpy = pc[HDIM * WDIM] - pry;
        const float dpz = pc[2 * HDIM * WDIM] - prz;
        const float ss   = fmaxf(dpx * dpx + dpy * dpy + dpz * dpz, 1e-12f);
        const float dist = sqrtf(ss);

        corr_s[i][r] = dot;
        geo_s[i][r]  = gamma * (-dist - 0.5f * fabsf(dpz));
    }
    __syncthreads();

    // Phase C: logits[e] = bv[e] + Wv[e,:] . corr + geo[e]
    if (r < DWIN) {
        float acc = bv[r];
        const float* wr = Wv + r * DWIN;
        for (int d = 0; d < DWIN; ++d) acc += wr[d] * corr_s[i][d];
        lg_s[i][r] = acc + geo_s[i][r];
    }
    __syncthreads();

    if (r == 0) {
        float mx = -3.4e38f;
        for (int d = 0; d < DWIN; ++d) mx = fmaxf(mx, lg_s[i][d]);
        mx_s[i] = mx;
    }
    __syncthreads();

    if (r < DWIN) lg_s[i][r] = __expf(lg_s[i][r] - mx_s[i]);
    __syncthreads();

    if (r == 0) {
        float sum = 0.f, sx = 0.f, sy = 0.f;
        for (int d = 0; d < DWIN; ++d) {
            const float e = lg_s[i][d];
            sum += e;
            sx  += e * (float)(d % 7 - RWIN);   // offx = dX
            sy  += e * (float)(d / 7 - RWIN);   // offy = dY
        }
        const float inv = 1.0f / sum;           // max weight = exp(0)/sum = inv
        out[pix]            = sx * inv;         // du
        out[NPIX + pix]     = sy * inv;         // dv
        out[2 * NPIX + pix] = inv;              // conf
    }
}

// ---------------------------------------------------------------------------
extern "C" void kernel_launch(void* const* d_in, const int* in_sizes, int n_in,
                              void* d_out, int out_size, void* d_ws, size_t ws_size,
                              hipStream_t stream)
{
    (void)in_sizes; (void)n_in; (void)out_size; (void)ws_size;
    const float* phi_cur = (const float*)d_in[0];
    const float* phi_rnd = (const float*)d_in[1];
    const float* P_cur   = (const float*)d_in[2];
    const float* P_rnd   = (const float*)d_in[3];
    const float* Wq      = (const float*)d_in[4];
    const float* Wk      = (const float*)d_in[5];
    const float* Wv      = (const float*)d_in[6];
    const float* bv      = (const float*)d_in[7];
    const float* gamma   = (const float*)d_in[8];

    float* qbuf = (float*)d_ws;                        // NPIX * 128 f32
    float* kbuf = qbuf + (size_t)NPIX * CMID;          // NPIX * 128 f32
    float* outp = (float*)d_out;

    dim3 g1(HDIM * WT, 2, 1);   // 784 x 2 blocks, 256 threads (8 waves)
    proj_norm_kernel<<<g1, 256, 0, stream>>>(phi_cur, Wq, qbuf);
    proj_norm_kernel<<<g1, 256, 0, stream>>>(phi_rnd, Wk, kbuf);

    const int nblocks = NPIX / 4;  // 6272
    corr_soft_kernel<<<nblocks, 256, 0, stream>>>(qbuf, kbuf, P_cur, P_rnd,
                                                  Wv, bv, gamma, outp);
}